// stuffRayMarcher_52149492908745
// MI455X (gfx1250) — compile-verified
//
#include <hip/hip_runtime.h>
#include <hip/hip_bf16.h>
#include <stdint.h>

// Stratified per-ray depth sampling for MI455X (gfx1250).
// Memory-bound (≈137 MB total traffic → ~6 us at 23.3 TB/s). No matrix math,
// so the CDNA5 paths exercised are:
//   * Tensor Data Mover: one tensor_load_to_lds per workgroup stages the 32KB
//     rand_u chunk with HARDWARE row padding (pad every 32 dwords by 1 dword ->
//     33-bank LDS stride, conflict-free per-ray reads), tracked by TENSORcnt.
//   * Async global->LDS B128 copies for the t0/t1 chunks, tracked by ASYNCcnt.
//   * gfx1250 split barriers / split wait counters.

#define NS            32
#define KBOX          4
#define SAMPLE_DEPTH  4.0f
#define BLOCK         256
#define RAYS_PER_B    (256 * 256)
#define NRAYS         (4 * RAYS_PER_B)
#define NOUT          ((size_t)NRAYS * (NS - 1))   // 8,126,464 per output tensor

// LDS layout: t0 block chunk (4KB) | t1 block chunk (4KB) | rand (256 rows x 33 dwords)
#define T0_OFF        0
#define T1_OFF        4096
#define RAND_OFF      8192
#define RAND_ROW_B    132                           // 32 floats + 1 pad dword -> stride 33 banks
#define SMEM_BYTES    (RAND_OFF + BLOCK * RAND_ROW_B)  // 41,984 B

typedef unsigned int v4u __attribute__((ext_vector_type(4)));
typedef int          v4i __attribute__((ext_vector_type(4)));
typedef int          v8i __attribute__((ext_vector_type(8)));

#if __has_builtin(__builtin_amdgcn_tensor_load_to_lds)
#define HAVE_TDM 1
#else
#define HAVE_TDM 0
#endif

__device__ __forceinline__ void async_copy_b128(uint32_t lds_byte_addr, const float* g) {
  // GLOBAL_LOAD_ASYNC_TO_LDS_B128: VDST = LDS byte address (VGPR),
  // VADDR = 64-bit global address (VGPR pair). Tracked by ASYNCcnt.
  asm volatile("global_load_async_to_lds_b128 %0, %1, off"
               :
               : "v"(lds_byte_addr), "v"((uint64_t)(uintptr_t)g)
               : "memory");
}

__device__ __forceinline__ void wait_async_zero() {
#if __has_builtin(__builtin_amdgcn_s_wait_asynccnt)
  __builtin_amdgcn_s_wait_asynccnt(0);
#else
  asm volatile("s_wait_asynccnt 0x0" ::: "memory");
#endif
}

__device__ __forceinline__ void wait_tensor_zero() {
#if __has_builtin(__builtin_amdgcn_s_wait_tensorcnt)
  __builtin_amdgcn_s_wait_tensorcnt(0);
#else
  asm volatile("s_wait_tensorcnt 0x0" ::: "memory");
#endif
}

__global__ __launch_bounds__(BLOCK) void raymarch_sample_kernel(
    const float* __restrict__ depth2,   // [bs,2,H,W,K]
    const float* __restrict__ rand_u,   // [bs,H,W,NS]
    float* __restrict__ out)            // rand_depth | new_dists | idx(as f32)
{
  __shared__ __align__(16) char smem[SMEM_BYTES];
  // Low 32 bits of the generic LDS address are the wave-relative LDS byte offset.
  const uint32_t lds_base = (uint32_t)(uintptr_t)&smem[0];

  const int tid  = threadIdx.x;
  const int ray0 = (int)blockIdx.x * BLOCK;          // block never crosses a batch (65536 % 256 == 0)
  const int b    = ray0 / RAYS_PER_B;
  const int hw0  = ray0 % RAYS_PER_B;

  const float* t0g = depth2 + ((size_t)(b * 2 + 0) * RAYS_PER_B + hw0) * KBOX;
  const float* t1g = depth2 + ((size_t)(b * 2 + 1) * RAYS_PER_B + hw0) * KBOX;
  const float* rg  = rand_u + (size_t)ray0 * NS;

  // ---- Stage t0/t1 via async global->LDS copies (one B128 per thread) ----
  async_copy_b128(lds_base + T0_OFF + tid * 16, t0g + tid * 4);
  async_copy_b128(lds_base + T1_OFF + tid * 16, t1g + tid * 4);

#if HAVE_TDM
  // ---- Stage the 32KB rand chunk with ONE Tensor Data Mover descriptor ----
  // 1D tile of 8192 dwords; hardware inserts 1 pad dword every 32 dwords
  // (pad_interval code 4 = 32 dwords, pad_amount code 0 = 1 dword), producing
  // the 33-dword LDS row stride directly.
  if (tid < 32) {                                    // only wave 0 issues the DMA
    const uint64_t ga = (uint64_t)(uintptr_t)rg;
    // D# group 0: count=1 | lds_addr | global_addr[56:0] | type=2
    const v4u g0 = { 1u,
                     lds_base + RAND_OFF,
                     (unsigned)(ga & 0xffffffffu),
                     (unsigned)((ga >> 32) & 0x01ffffffu) | (2u << 30) };
    // D# group 1:
    // w0: wg_mask=0 | data_size=2(4B) | pad_enable=1 | pad_interval=4 | pad_amount=0
    // w1[31:16]=tensor_dim0.lo=8192   w2[31:16]=tensor_dim1.lo=1
    // w3[31:16]=tile_dim0=8192        w4[15:0]=tile_dim1=1, tile_dim2=0
    // w5=tensor_dim0_stride=8192      w6,w7=0
    const v8i g1 = { (int)((2u << 16) | (1u << 20) | (4u << 22)),
                     (int)(0x2000u << 16),
                     (int)(1u << 16),
                     (int)(0x2000u << 16),
                     1,
                     0x2000,
                     0,
                     0 };
    const v4i g2 = { 0, 0, 0, 0 };
    const v4i g3 = { 0, 0, 0, 0 };
#if defined(__clang_major__) && (__clang_major__ >= 23)
    const v8i g4 = { 0, 0, 0, 0, 0, 0, 0, 0 };
    __builtin_amdgcn_tensor_load_to_lds(g0, g1, g2, g3, g4, 0);
#else
    __builtin_amdgcn_tensor_load_to_lds(g0, g1, g2, g3, 0);
#endif
    wait_tensor_zero();                              // TENSORcnt -> 0 before barrier
  }
#else
  // Fallback: per-lane async B128 staging of the rand chunk (dense 512B/instr).
#pragma unroll
  for (int j = 0; j < 8; ++j) {
    const int q4 = tid + j * BLOCK;                  // vec4 index into the 8K-float chunk
    const uint32_t dst = lds_base + RAND_OFF +
                         (uint32_t)(q4 >> 3) * RAND_ROW_B + (uint32_t)(q4 & 7) * 16;
    async_copy_b128(dst, rg + (size_t)q4 * 4);
  }
#endif

  wait_async_zero();      // my wave's async copies done
  __syncthreads();        // all waves' staged data visible

  // ---- Per-ray scalar pipeline ----
  const float4 t0v = *(const float4*)(smem + T0_OFF + tid * 16);
  const float4 t1v = *(const float4*)(smem + T1_OFF + tid * 16);

  float d0 = t1v.x - t0v.x, d1 = t1v.y - t0v.y, d2 = t1v.z - t0v.z, d3 = t1v.w - t0v.w;
  d0 = (d0 != d0) ? 0.0f : d0;                       // NaN (invalid box) -> 0
  d1 = (d1 != d1) ? 0.0f : d1;
  d2 = (d2 != d2) ? 0.0f : d2;
  d3 = (d3 != d3) ? 0.0f : d3;

  const float a0 = d0, a1 = a0 + d1, a2 = a1 + d2, a3 = a2 + d3;   // accu_depth
  const float total = fminf(a3, SAMPLE_DEPTH);
  const float scale = total * (1.0f / (float)NS);

  // depth_deltas prefix (NaN entries for invalid boxes are never selected: idx <= nbox-1)
  const float e0 = t0v.x;
  const float e1 = e0 + (t0v.y - t1v.x);
  const float e2 = e1 + (t0v.z - t1v.y);
  const float e3 = e2 + (t0v.w - t1v.z);

  const float* rrow = (const float*)(smem + RAND_OFF + (size_t)tid * RAND_ROW_B);

  float* out_depth = out;
  float* out_dists = out + NOUT;
  float* out_idx   = out + 2 * NOUT;
  const size_t obase = (size_t)(ray0 + tid) * (NS - 1);

  // strat is already monotone and total >= 0 -> reference's sort is identity.
  float prev = rrow[0] * scale;                      // (rand[0] + 0) * total/32
#pragma unroll
  for (int i = 1; i < NS; ++i) {
    const float cur = (rrow[i] + (float)i) * scale;
    const float mid = 0.5f * (cur + prev);
    const float nd  = cur - prev;

    int cnt = (mid > a0) + (mid > a1) + (mid > a2) + (mid > a3);
    const int ci = cnt < 3 ? cnt : 3;                // take_along_axis clips
    const float head = (ci == 0) ? e0 : (ci == 1) ? e1 : (ci == 2) ? e2 : e3;

    out_depth[obase + (i - 1)] = head + mid;
    out_dists[obase + (i - 1)] = nd;
    out_idx  [obase + (i - 1)] = (float)cnt;
    prev = cur;
  }
}

extern "C" void kernel_launch(void* const* d_in, const int* in_sizes, int n_in,
                              void* d_out, int out_size, void* d_ws, size_t ws_size,
                              hipStream_t stream) {
  (void)n_in; (void)out_size; (void)d_ws; (void)ws_size;
  const float* depth2 = (const float*)d_in[0];
  const float* rand_u = (const float*)d_in[1];
  float* out = (float*)d_out;

  const int nrays = in_sizes[1] / NS;                // 262,144
  dim3 grid((nrays + BLOCK - 1) / BLOCK);            // 1024 blocks of 256 (8 waves)
  hipLaunchKernelGGL(raymarch_sample_kernel, grid, dim3(BLOCK), 0, stream,
                     depth2, rand_u, out);
}